// ReluSegmentNetwork2D_44014824849725
// MI455X (gfx1250) — compile-verified
//
#include <hip/hip_runtime.h>

// CDNA5 / gfx1250: wave32, WMMA f32 16x16x4.
typedef __attribute__((ext_vector_type(2))) float v2f;
typedef __attribute__((ext_vector_type(8))) float v8f;

// Reformulated so the point-dependent contraction (over j) happens inside WMMA:
//   G1[i,p] = sum_j z00[i,j]*c_j(y_p) + z01[i,j]*d_j(y_p)   (A = Z view, B = c/d weights)
//   G2[i,p] = sum_j z10[i,j]*c_j(y_p) + z11[i,j]*d_j(y_p)
//   out[p]  = sum_i a_i(x_p)*G1[i,p] + b_i(x_p)*G2[i,p]     (in-lane over 8 acc VGPRs,
//                                                            + one xor-16 shuffle)
// A-matrix (16x4 f32): lane L holds row M = i = L%16; VGPR v covers K = j = 2*(L/16)+v.
// B-matrix (4x16 f32): lane L holds col N = p = L%16; VGPR v covers K = j = 2*(L/16)+v.
// C/D  (16x16 f32):    VGPR r, lane L -> (i = r + 8*(L/16), p = L%16).
__global__ void __launch_bounds__(256)
relu_seg2d_wmma_kernel(const float* __restrict__ x,
                       const float* __restrict__ y,
                       const float* __restrict__ zg,   // 17x17 row-major
                       float* __restrict__ out,
                       int ntiles)
{
    const int lane = threadIdx.x & 31;
    const int wave = (int)((blockIdx.x * blockDim.x + threadIdx.x) >> 5);
    const int nwaves = (int)((gridDim.x * blockDim.x) >> 5);
    const int half = lane >> 4;      // 0: lanes 0-15, 1: lanes 16-31
    const int col  = lane & 15;      // point index within tile (N) / row i (A operand)

    // Loop-invariant A fragments: four shifted 16x16 views of z_grid in A-layout.
    // Za00[i][j]=zg[i][j]  Za10=zg[i+1][j]  Za01=zg[i][j+1]  Za11=zg[i+1][j+1]
    v2f Za00[4], Za10[4], Za01[4], Za11[4];
#pragma unroll
    for (int kk = 0; kk < 4; ++kk) {
#pragma unroll
        for (int v = 0; v < 2; ++v) {
            const int i = col;                     // A row (M)
            const int j = 4 * kk + 2 * half + v;   // A column (K)
            Za00[kk][v] = zg[(i)     * 17 + (j)    ];
            Za10[kk][v] = zg[(i + 1) * 17 + (j)    ];
            Za01[kk][v] = zg[(i)     * 17 + (j + 1)];
            Za11[kk][v] = zg[(i + 1) * 17 + (j + 1)];
        }
    }

    for (int tile = wave; tile < ntiles; tile += nwaves) {
        const int tb = tile * 16;
        const float xv = x[tb + col];   // this lane's point's x
        const float yv = y[tb + col];   // this lane's point's y

        // Prefetch next tile's x/y (speculative; dropped if OOB).
        const int ntb = tb + nwaves * 16;
        __builtin_prefetch(&x[ntb + col], 0, 1);
        __builtin_prefetch(&y[ntb + col], 0, 1);

        // B fragments: per-point y weights, c_j = clamp(y-j,0,1)*(1-(y-j)),
        //                                   d_j = clamp(y-j,0,1)*(y-j)
        v2f Cw[4], Dw[4];
#pragma unroll
        for (int kk = 0; kk < 4; ++kk) {
#pragma unroll
            for (int v = 0; v < 2; ++v) {
                const float fj = (float)(4 * kk + 2 * half + v);
                const float t  = yv - fj;
                const float ch = fminf(fmaxf(t, 0.0f), 1.0f);
                Cw[kk][v] = ch * (1.0f - t);
                Dw[kk][v] = ch * t;
            }
        }

        // G1 = Z00@Cw + Z01@Dw ; G2 = Z10@Cw + Z11@Dw   (K=16 via 4 chunks of 4)
        v8f g1 = {};
        v8f g2 = {};
#pragma unroll
        for (int kk = 0; kk < 4; ++kk) {
            g1 = __builtin_amdgcn_wmma_f32_16x16x4_f32(false, Za00[kk], false, Cw[kk],
                                                       (short)0, g1, false, false);
            g2 = __builtin_amdgcn_wmma_f32_16x16x4_f32(false, Za10[kk], false, Cw[kk],
                                                       (short)0, g2, false, false);
            g1 = __builtin_amdgcn_wmma_f32_16x16x4_f32(false, Za01[kk], false, Dw[kk],
                                                       (short)0, g1, false, false);
            g2 = __builtin_amdgcn_wmma_f32_16x16x4_f32(false, Za11[kk], false, Dw[kk],
                                                       (short)0, g2, false, false);
        }

        // out[p] = sum_i a_i(x_p)*G1[i,p] + b_i(x_p)*G2[i,p]
        // Lane L holds i = r + 8*half for r=0..7 -> in-lane partial over half the i range.
        float s = 0.0f;
#pragma unroll
        for (int r = 0; r < 8; ++r) {
            const float fi = (float)(r + 8 * half);
            const float t  = xv - fi;
            const float ch = fminf(fmaxf(t, 0.0f), 1.0f);
            const float a  = ch * (1.0f - t);
            const float b  = ch * t;
            s += a * g1[r] + b * g2[r];
        }
        // Combine the two i-halves (lanes L and L+16 hold the same point p=col).
        s += __shfl_xor(s, 16, 32);
        if (half == 0) out[tb + col] = s;   // one coalesced 64B store
    }
}

extern "C" void kernel_launch(void* const* d_in, const int* in_sizes, int n_in,
                              void* d_out, int out_size, void* d_ws, size_t ws_size,
                              hipStream_t stream) {
    // inputs (setup_inputs order): x[P], y[P], x_points[17], y_points[17], z_grid[17*17]
    const float* x  = (const float*)d_in[0];
    const float* y  = (const float*)d_in[1];
    const float* zg = (const float*)d_in[4];
    float* out = (float*)d_out;

    const int P = in_sizes[0];          // 4194304
    const int ntiles = P / 16;          // 262144 (P divisible by 16)

    dim3 grid(2048), block(256);        // 16384 wave32s -> 16 tiles/wave
    relu_seg2d_wmma_kernel<<<grid, block, 0, stream>>>(x, y, zg, out, ntiles);
}